// DifferentiableLinearSolver_249108103947
// MI455X (gfx1250) — compile-verified
//
#include <hip/hip_runtime.h>

// Conjugate gradient for 8192x8192 fp32 SPD system, 50 iterations.
// Matvec is HBM-bandwidth bound (A = 268 MB > 192 MB L2, streamed every
// iteration -> ~575 us floor at 23.3 TB/s). fp32 fidelity kept by running the
// matvec through V_WMMA_F32_16X16X4_F32 (exact f32 multiply/accumulate).
// A-loads are non-temporal (zero reuse, would only thrash L2); p stays cached.
// The matvec k-loop is explicitly double-buffered so the next 8 loads are in
// flight while the current WMMA quad executes (sustained per-wave MLP ~8-16,
// x 8192 waves -> enough bytes in flight to cover HBM latency at 23.3 TB/s).
// All scalar/vector CG bookkeeping is fused into ONE single-workgroup kernel
// per iteration (alpha/beta live in LDS, never round-trip through HBM).

#define NDIM 8192
#define SEGS 16                // K-segments per row-block -> 8192 waves in matvec
#define SEGK (NDIM / SEGS)     // 512
#define MAX_ITER 50

typedef __attribute__((ext_vector_type(2))) float v2f;
typedef __attribute__((ext_vector_type(8))) float v8f;

// workspace layout (floats)
#define WS_APPART 0                     // SEGS * NDIM partial matvec results
#define WS_R      (SEGS * NDIM)         // NDIM
#define WS_P      (WS_R + NDIM)         // NDIM
#define WS_SCAL   (WS_P + NDIM)         // scalars
#define S_RSOLD 0
#define S_DONE  1

#define ELEMS_PER_THREAD (NDIM / 1024)  // 8

#define WMMA_F32(a, b, c) \
  __builtin_amdgcn_wmma_f32_16x16x4_f32(false, (a), false, (b), (short)0, (c), false, false)

// ---------------- init ----------------
__global__ void cg_init_vec(const float* __restrict__ b, float* __restrict__ x,
                            float* __restrict__ r, float* __restrict__ p) {
  int i = blockIdx.x * blockDim.x + threadIdx.x;
  float bi = b[i];
  x[i] = 0.0f;   // x0 = 0  -> r = b, p = b
  r[i] = bi;
  p[i] = bi;
}

__global__ __launch_bounds__(1024) void cg_init_scal(const float* __restrict__ b,
                                                     float* __restrict__ scal) {
  __shared__ float sm[1024];
  float acc = 0.0f;
  for (int i = threadIdx.x; i < NDIM; i += 1024) { float v = b[i]; acc += v * v; }
  sm[threadIdx.x] = acc;
  __syncthreads();
  for (int off = 512; off > 0; off >>= 1) {
    if (threadIdx.x < (unsigned)off) sm[threadIdx.x] += sm[threadIdx.x + off];
    __syncthreads();
  }
  if (threadIdx.x == 0) {
    scal[S_RSOLD] = sm[0];
    scal[S_DONE]  = 0.0f;
  }
}

// ---------------- matvec via WMMA f32 16x16x4 ----------------
// Wave w: row-block rb = w/SEGS (16 rows), K-segment seg = w%SEGS (512 cols).
// A operand (16x4 f32, 2 VGPRs): lanes 0-15 carry K=k,k+1 of row M=lane;
// lanes 16-31 carry K=k+2,k+3 of row M=lane-16  -> one b64 NT load per lane.
// B operand: p[k..k+3] replicated into all 16 columns (high half-lanes offset
// by half the K dimension, matching the ISA striping pattern), so every column
// of D equals the matvec partial; lane 0 holds rows 0-7, lane 16 rows 8-15.
__global__ __launch_bounds__(256) void cg_matvec_wmma(const float* __restrict__ A,
                                                      const float* __restrict__ p,
                                                      float* __restrict__ ap_part) {
  const int wave = (int)((blockIdx.x * blockDim.x + threadIdx.x) >> 5);
  const int lane = threadIdx.x & 31;
  const int rb   = wave >> 4;          // wave / SEGS
  const int seg  = wave & (SEGS - 1);  // wave % SEGS
  const int m    = lane & 15;
  const int kk   = (lane >> 4) << 1;   // 0 for lanes 0-15, 2 for lanes 16-31

  const float* __restrict__ arow = A + (size_t)(rb * 16 + m) * NDIM + kk;
  const float* __restrict__ pv   = p + kk;
  const int kbase = seg * SEGK;
  const int kend  = kbase + SEGK;

  v8f c0 = {}, c1 = {}, c2 = {}, c3 = {};

  // prologue: load group at kbase
  v2f a0 = __builtin_nontemporal_load((const v2f*)(arow + kbase));
  v2f a1 = __builtin_nontemporal_load((const v2f*)(arow + kbase + 4));
  v2f a2 = __builtin_nontemporal_load((const v2f*)(arow + kbase + 8));
  v2f a3 = __builtin_nontemporal_load((const v2f*)(arow + kbase + 12));
  v2f b0 = *(const v2f*)(pv + kbase);
  v2f b1 = *(const v2f*)(pv + kbase + 4);
  v2f b2 = *(const v2f*)(pv + kbase + 8);
  v2f b3 = *(const v2f*)(pv + kbase + 12);

  for (int k = kbase; k < kend; k += 16) {
    // prefetch NEXT group before consuming the current one (keeps ~8-16 loads
    // in flight per wave). Final iteration wraps to kbase: unconditional,
    // branch-free, cache-hit, results discarded.
    const int kn = (k + 16 < kend) ? (k + 16) : kbase;
    v2f na0 = __builtin_nontemporal_load((const v2f*)(arow + kn));
    v2f na1 = __builtin_nontemporal_load((const v2f*)(arow + kn + 4));
    v2f na2 = __builtin_nontemporal_load((const v2f*)(arow + kn + 8));
    v2f na3 = __builtin_nontemporal_load((const v2f*)(arow + kn + 12));
    v2f nb0 = *(const v2f*)(pv + kn);
    v2f nb1 = *(const v2f*)(pv + kn + 4);
    v2f nb2 = *(const v2f*)(pv + kn + 8);
    v2f nb3 = *(const v2f*)(pv + kn + 12);

    c0 = WMMA_F32(a0, b0, c0);
    c1 = WMMA_F32(a1, b1, c1);
    c2 = WMMA_F32(a2, b2, c2);
    c3 = WMMA_F32(a3, b3, c3);

    a0 = na0; a1 = na1; a2 = na2; a3 = na3;
    b0 = nb0; b1 = nb1; b2 = nb2; b3 = nb3;
  }
  v8f c = (c0 + c1) + (c2 + c3);

  if (m == 0) {  // lanes 0 and 16: write rows rb*16..+7 and rb*16+8..+15
    float* out = ap_part + (size_t)seg * NDIM + rb * 16 + ((lane >> 4) << 3);
    *(v8f*)out = c;
  }
}

// ---------------- fused CG bookkeeping (single workgroup) ----------------
// Phase A: Ap[i] = sum_seg partials + 1e-6*p[i] (in registers), pAp -> alpha
// Phase B: x += alpha p ; r -= alpha Ap (freeze-aware) ; rsnew -> beta, flags
// Phase C: p = frozen ? p : r + beta*p
__global__ __launch_bounds__(1024) void cg_update(const float* __restrict__ ap_part,
                                                  float* __restrict__ x,
                                                  float* __restrict__ r,
                                                  float* __restrict__ p,
                                                  float* __restrict__ scal) {
  __shared__ float red[1024];
  __shared__ float sh_alpha, sh_beta, sh_frozen;

  const unsigned tid = threadIdx.x;
  const float done = scal[S_DONE];      // read by all threads before any write

  float api[ELEMS_PER_THREAD];
  float pi[ELEMS_PER_THREAD];
  float ri[ELEMS_PER_THREAD];

  // ---- Phase A: assemble Ap, reduce p.Ap ----
  float acc = 0.0f;
#pragma unroll
  for (int t = 0; t < ELEMS_PER_THREAD; ++t) {
    const int i = tid + t * 1024;
    const float pv = p[i];
    float s = 1e-6f * pv;               // A_reg = A + 1e-6 I
#pragma unroll
    for (int g = 0; g < SEGS; ++g) s += ap_part[g * NDIM + i];
    pi[t]  = pv;
    api[t] = s;
    acc   += pv * s;
  }
  red[tid] = acc;
  __syncthreads();
  for (int off = 512; off > 0; off >>= 1) {
    if (tid < (unsigned)off) red[tid] += red[tid + off];
    __syncthreads();
  }
  if (tid == 0) sh_alpha = scal[S_RSOLD] / (red[0] + 1e-12f);
  __syncthreads();
  const float alpha = sh_alpha;

  // ---- Phase B: x/r update (freeze-aware), reduce rsnew ----
  acc = 0.0f;
#pragma unroll
  for (int t = 0; t < ELEMS_PER_THREAD; ++t) {
    const int i = tid + t * 1024;
    float rv = r[i];
    if (done == 0.0f) {
      x[i] += alpha * pi[t];
      rv   -= alpha * api[t];
      r[i]  = rv;
    }
    ri[t] = rv;
    acc  += rv * rv;
  }
  __syncthreads();                       // red[] reuse
  red[tid] = acc;
  __syncthreads();
  for (int off = 512; off > 0; off >>= 1) {
    if (tid < (unsigned)off) red[tid] += red[tid + off];
    __syncthreads();
  }
  if (tid == 0) {
    const float rsnew = red[0];
    const float rsold = scal[S_RSOLD];
    const bool  conv  = sqrtf(rsnew) < 1e-6f;
    const float frozen = (done != 0.0f || conv) ? 1.0f : 0.0f;
    sh_beta   = rsnew / (rsold + 1e-12f);
    sh_frozen = frozen;
    if (frozen == 0.0f) scal[S_RSOLD] = rsnew;  // rs2 = frozen ? rsold : rsnew
    scal[S_DONE] = frozen;                      // next iteration's done
  }
  __syncthreads();

  // ---- Phase C: p update ----
  if (sh_frozen == 0.0f) {
    const float beta = sh_beta;
#pragma unroll
    for (int t = 0; t < ELEMS_PER_THREAD; ++t) {
      const int i = tid + t * 1024;
      p[i] = ri[t] + beta * pi[t];
    }
  }
}

extern "C" void kernel_launch(void* const* d_in, const int* in_sizes, int n_in,
                              void* d_out, int out_size, void* d_ws, size_t ws_size,
                              hipStream_t stream) {
  (void)in_sizes; (void)n_in; (void)out_size; (void)ws_size;
  const float* A = (const float*)d_in[0];
  const float* b = (const float*)d_in[1];
  float* x  = (float*)d_out;            // solution lives directly in d_out
  float* ws = (float*)d_ws;             // needs ~590 KB

  float* ap_part = ws + WS_APPART;
  float* r       = ws + WS_R;
  float* p       = ws + WS_P;
  float* scal    = ws + WS_SCAL;

  cg_init_vec<<<NDIM / 256, 256, 0, stream>>>(b, x, r, p);
  cg_init_scal<<<1, 1024, 0, stream>>>(b, scal);

  const int mv_blocks = ((NDIM / 16) * SEGS * 32) / 256;  // 8192 waves -> 1024 blocks
  for (int it = 0; it < MAX_ITER; ++it) {
    cg_matvec_wmma<<<mv_blocks, 256, 0, stream>>>(A, p, ap_part);
    cg_update<<<1, 1024, 0, stream>>>(ap_part, x, r, p, scal);
  }
}